// MetabolicFinetuneModel_90031104458777
// MI455X (gfx1250) — compile-verified
//
#include <hip/hip_runtime.h>
#include <stdint.h>
#include <math.h>

#ifndef __has_builtin
#define __has_builtin(x) 0
#endif

#define NPOINTS 128
#define FLOATS_PER_SAMPLE (NPOINTS * 3)   // 384 floats = 1536 B
#define LANES_PER_SAMPLE 4
#define SAMPLES_PER_WAVE 8
#define WAVES_PER_BLOCK 2                 // 64 threads
#define SAMPLES_PER_BLOCK (WAVES_PER_BLOCK * SAMPLES_PER_WAVE)  // 16
#define SAMPLE_STRIDE_F 388               // 1552 B: +16B pad -> optimal LDS banking
#define Q_OFF_F (SAMPLES_PER_WAVE * SAMPLE_STRIDE_F)            // 3104 floats
#define WAVE_FLOATS (2 * SAMPLES_PER_WAVE * SAMPLE_STRIDE_F)    // 6208 floats

typedef int v4i_vs __attribute__((vector_size(16)));            // matches builtin param
typedef __attribute__((address_space(3))) v4i_vs* lds_v4i_ptr;  // LDS-side pointer

// Sum across a 4-lane group (lanes sharing lane>>2): 2 butterfly steps.
__device__ __forceinline__ float group_sum(float v) {
  v += __shfl_xor(v, 2, 32);
  v += __shfl_xor(v, 1, 32);
  return v;
}

__global__ __launch_bounds__(64) void kabsch_mse_kernel(
    const float* __restrict__ P, const float* __restrict__ Q,
    const unsigned char* __restrict__ pad, float* __restrict__ block_sums,
    int B) {
  __shared__ __align__(16) float smem[WAVES_PER_BLOCK * WAVE_FLOATS];  // ~48.5 KB
  __shared__ float shLoss[SAMPLES_PER_BLOCK];
  const int lane = threadIdx.x & 31;
  const int wv = threadIdx.x >> 5;
  const int grp = lane >> 2;   // sample slot within wave (0..7)
  const int lig = lane & 3;    // lane within 4-lane group

  const int origBase = blockIdx.x * SAMPLES_PER_BLOCK + wv * SAMPLES_PER_WAVE;
  int waveBase = origBase;
  if (waveBase + SAMPLES_PER_WAVE > B) {               // clamp (B=32768 -> never)
    waveBase = (B >= SAMPLES_PER_WAVE) ? (B - SAMPLES_PER_WAVE) : 0;
  }
  const int sample = waveBase + grp;

  float* sw = &smem[wv * WAVE_FLOATS];
  const float* gp = P + (size_t)waveBase * FLOATS_PER_SAMPLE;  // 12288 contiguous B
  const float* gq = Q + (size_t)waveBase * FLOATS_PER_SAMPLE;

  // ---- Async stage this wave's 8 samples (P and Q) into padded LDS ----
#if __has_builtin(__builtin_amdgcn_global_load_async_to_lds_b128)
#pragma unroll
  for (int s = 0; s < SAMPLES_PER_WAVE; ++s) {
    const float* srcP = gp + s * FLOATS_PER_SAMPLE;
    const float* srcQ = gq + s * FLOATS_PER_SAMPLE;
    float* dstP = sw + s * SAMPLE_STRIDE_F;
    float* dstQ = sw + Q_OFF_F + s * SAMPLE_STRIDE_F;
#pragma unroll
    for (int c = 0; c < 3; ++c) {   // 3 * 512 B per sample per matrix
      __builtin_amdgcn_global_load_async_to_lds_b128(
          (v4i_vs*)(srcP + c * 128 + lane * 4),
          (lds_v4i_ptr)(dstP + c * 128 + lane * 4), 0, 0);
      __builtin_amdgcn_global_load_async_to_lds_b128(
          (v4i_vs*)(srcQ + c * 128 + lane * 4),
          (lds_v4i_ptr)(dstQ + c * 128 + lane * 4), 0, 0);
    }
  }
  asm volatile("s_wait_asynccnt 0" ::: "memory");
#else
#pragma unroll
  for (int s = 0; s < SAMPLES_PER_WAVE; ++s) {
#pragma unroll
    for (int c = 0; c < 3; ++c) {
      float4 a = ((const float4*)(gp + s * FLOATS_PER_SAMPLE))[c * 32 + lane];
      float4 b = ((const float4*)(gq + s * FLOATS_PER_SAMPLE))[c * 32 + lane];
      ((float4*)(sw + s * SAMPLE_STRIDE_F))[c * 32 + lane] = a;
      ((float4*)(sw + Q_OFF_F + s * SAMPLE_STRIDE_F))[c * 32 + lane] = b;
    }
  }
#endif
  // Wave-private LDS region: no cross-wave barrier needed for data staging.

  // Each lane owns 32 consecutive points (96 floats = 384 B) of its sample.
  const float* myP = sw + grp * SAMPLE_STRIDE_F + lig * 96;
  const float* myQ = sw + Q_OFF_F + grp * SAMPLE_STRIDE_F + lig * 96;
  const uint4 mr0 = *(const uint4*)(pad + (size_t)sample * NPOINTS + lig * 32);
  const uint4 mr1 = *(const uint4*)(pad + (size_t)sample * NPOINTS + lig * 32 + 16);
  const unsigned int mword[8] = {mr0.x, mr0.y, mr0.z, mr0.w,
                                 mr1.x, mr1.y, mr1.z, mr1.w};

  // ---- Single pass: raw weighted moments (18 accumulators) ----
  float cnt = 0.f;
  float sPx = 0.f, sPy = 0.f, sPz = 0.f, sQx = 0.f, sQy = 0.f, sQz = 0.f;
  float m00 = 0.f, m01 = 0.f, m02 = 0.f, m10 = 0.f, m11 = 0.f, m12 = 0.f,
        m20 = 0.f, m21 = 0.f, m22 = 0.f;
  float sp2 = 0.f, sq2 = 0.f;

#pragma unroll
  for (int c = 0; c < 8; ++c) {    // 8 chunks of 4 points (12 floats = 3 float4)
    const float4 f0 = ((const float4*)myP)[c * 3 + 0];
    const float4 f1 = ((const float4*)myP)[c * 3 + 1];
    const float4 f2 = ((const float4*)myP)[c * 3 + 2];
    const float4 g0 = ((const float4*)myQ)[c * 3 + 0];
    const float4 g1 = ((const float4*)myQ)[c * 3 + 1];
    const float4 g2 = ((const float4*)myQ)[c * 3 + 2];
    const float pxv[4] = {f0.x, f0.w, f1.z, f2.y};
    const float pyv[4] = {f0.y, f1.x, f1.w, f2.z};
    const float pzv[4] = {f0.z, f1.y, f2.x, f2.w};
    const float qxv[4] = {g0.x, g0.w, g1.z, g2.y};
    const float qyv[4] = {g0.y, g1.x, g1.w, g2.z};
    const float qzv[4] = {g0.z, g1.y, g2.x, g2.w};
    const unsigned int mw = mword[c];
#pragma unroll
    for (int k = 0; k < 4; ++k) {
      const float w = (((mw >> (8 * k)) & 0xFFu) == 0u) ? 1.0f : 0.0f;
      const float px = pxv[k], py = pyv[k], pz = pzv[k];
      const float qx = qxv[k], qy = qyv[k], qz = qzv[k];
      cnt += w;
      const float wpx = w * px, wpy = w * py, wpz = w * pz;
      sPx += wpx; sPy += wpy; sPz += wpz;
      sQx = fmaf(w, qx, sQx); sQy = fmaf(w, qy, sQy); sQz = fmaf(w, qz, sQz);
      m00 = fmaf(wpx, qx, m00); m01 = fmaf(wpx, qy, m01); m02 = fmaf(wpx, qz, m02);
      m10 = fmaf(wpy, qx, m10); m11 = fmaf(wpy, qy, m11); m12 = fmaf(wpy, qz, m12);
      m20 = fmaf(wpz, qx, m20); m21 = fmaf(wpz, qy, m21); m22 = fmaf(wpz, qz, m22);
      sp2 = fmaf(wpx, px, fmaf(wpy, py, fmaf(wpz, pz, sp2)));
      sq2 = fmaf(w, qx * qx + qy * qy + qz * qz, sq2);
    }
  }

  // ---- Reduce 18 values across the 4-lane group ----
  cnt = group_sum(cnt);
  sPx = group_sum(sPx); sPy = group_sum(sPy); sPz = group_sum(sPz);
  sQx = group_sum(sQx); sQy = group_sum(sQy); sQz = group_sum(sQz);
  m00 = group_sum(m00); m01 = group_sum(m01); m02 = group_sum(m02);
  m10 = group_sum(m10); m11 = group_sum(m11); m12 = group_sum(m12);
  m20 = group_sum(m20); m21 = group_sum(m21); m22 = group_sum(m22);
  sp2 = group_sum(sp2); sq2 = group_sum(sq2);

  // Plain MSE numerator: sum w|p-q|^2 = sp2 + sq2 - 2*trace(M)
  const float plain = sp2 + sq2 - 2.0f * (m00 + m11 + m22);

  // ---- Centered quantities via covariance identity ----
  const float inv_n = (cnt > 0.f) ? (1.0f / cnt) : 0.f;
  const float h00 = m00 - sPx * sQx * inv_n, h01 = m01 - sPx * sQy * inv_n,
              h02 = m02 - sPx * sQz * inv_n;
  const float h10 = m10 - sPy * sQx * inv_n, h11 = m11 - sPy * sQy * inv_n,
              h12 = m12 - sPy * sQz * inv_n;
  const float h20 = m20 - sPz * sQx * inv_n, h21 = m21 - sPz * sQy * inv_n,
              h22 = m22 - sPz * sQz * inv_n;
  const float nP = sp2 - (sPx * sPx + sPy * sPy + sPz * sPz) * inv_n;
  const float nQ = sq2 - (sQx * sQx + sQy * sQy + sQz * sQz) * inv_n;

  // ---- Singular values of H via eigenvalues of A = H^T H (closed form) ----
  const float a00 = h00 * h00 + h10 * h10 + h20 * h20;
  const float a11 = h01 * h01 + h11 * h11 + h21 * h21;
  const float a22 = h02 * h02 + h12 * h12 + h22 * h22;
  const float a01 = h00 * h01 + h10 * h11 + h20 * h21;
  const float a02 = h00 * h02 + h10 * h12 + h20 * h22;
  const float a12 = h01 * h02 + h11 * h12 + h21 * h22;

  const float qm = (a00 + a11 + a22) * (1.0f / 3.0f);
  const float p1 = a01 * a01 + a02 * a02 + a12 * a12;
  const float b00 = a00 - qm, b11 = a11 - qm, b22 = a22 - qm;
  const float p2 = b00 * b00 + b11 * b11 + b22 * b22 + 2.0f * p1;
  float e1, e2, e3;
  if (p2 <= 1e-30f) {
    e1 = e2 = e3 = qm;
  } else {
    const float pp = sqrtf(p2 * (1.0f / 6.0f));
    const float ip = 1.0f / pp;
    const float c00 = b00 * ip, c11 = b11 * ip, c22 = b22 * ip;
    const float c01 = a01 * ip, c02 = a02 * ip, c12 = a12 * ip;
    float r = 0.5f * (c00 * (c11 * c22 - c12 * c12) -
                      c01 * (c01 * c22 - c12 * c02) +
                      c02 * (c01 * c12 - c11 * c02));
    r = fminf(1.0f, fmaxf(-1.0f, r));
    const float phi = acosf(r) * (1.0f / 3.0f);
    e1 = qm + 2.0f * pp * cosf(phi);
    e3 = qm + 2.0f * pp * cosf(phi + 2.0943951023931953f);  // +2*pi/3
    e2 = 3.0f * qm - e1 - e3;
  }
  const float s1 = sqrtf(fmaxf(e1, 0.f));
  const float s2 = sqrtf(fmaxf(e2, 0.f));
  const float s3 = sqrtf(fmaxf(e3, 0.f));  // smallest singular value

  const float detH = h00 * (h11 * h22 - h12 * h21) -
                     h01 * (h10 * h22 - h12 * h20) +
                     h02 * (h10 * h21 - h11 * h20);
  const float dsgn = (detH > 0.f) ? 1.f : ((detH < 0.f) ? -1.f : 0.f);
  const float corr = s1 + s2 + dsgn * s3;

  const float denom = 1.0f / (3.0f * fmaxf(cnt, 1.0f));
  const float kab = fmaxf(nP + nQ - 2.0f * corr, 0.0f) * denom;
  const float pmse = fmaxf(plain, 0.0f) * denom;
  const float res = (cnt >= 3.0f) ? kab : pmse;

  // ---- Deterministic per-block partial sum (canonical ownership only) ----
  const int origSample = origBase + grp;
  if (lig == 0) shLoss[wv * SAMPLES_PER_WAVE + grp] = (origSample < B) ? res : 0.0f;
  __syncthreads();
  if (threadIdx.x == 0) {
    float s = 0.f;
#pragma unroll
    for (int i = 0; i < SAMPLES_PER_BLOCK; ++i) s += shLoss[i];
    block_sums[blockIdx.x] = s;
  }
}

// Deterministic single-block mean over block partial sums (fixed order).
__global__ __launch_bounds__(256) void reduce_mean_kernel(
    const float* __restrict__ block_sums, float* __restrict__ out, int nblocks,
    float invB) {
  __shared__ float sh[256];
  const int t = threadIdx.x;
  float s = 0.f;
  for (int i = t; i < nblocks; i += 256) s += block_sums[i];
  sh[t] = s;
  __syncthreads();
  for (int stride = 128; stride > 0; stride >>= 1) {
    if (t < stride) sh[t] += sh[t + stride];
    __syncthreads();
  }
  if (t == 0) out[0] = sh[0] * invB;
}

extern "C" void kernel_launch(void* const* d_in, const int* in_sizes, int n_in,
                              void* d_out, int out_size, void* d_ws, size_t ws_size,
                              hipStream_t stream) {
  const float* P = (const float*)d_in[0];
  const float* Q = (const float*)d_in[1];
  const unsigned char* pad = (const unsigned char*)d_in[2];
  float* out = (float*)d_out;
  float* ws = (float*)d_ws;  // nblocks floats of per-block partial sums

  const int B = in_sizes[0] / FLOATS_PER_SAMPLE;
  const int blocks = (B + SAMPLES_PER_BLOCK - 1) / SAMPLES_PER_BLOCK;

  kabsch_mse_kernel<<<blocks, 64, 0, stream>>>(P, Q, pad, ws, B);
  reduce_mean_kernel<<<1, 256, 0, stream>>>(ws, out, blocks, 1.0f / (float)B);
}